// LinearDynamicsExact_10892037062791
// MI455X (gfx1250) — compile-verified
//
#include <hip/hip_runtime.h>
#include <math.h>

// ---------------------------------------------------------------------------
// LinearDynamicsExact on gfx1250: diagonalized complex scan + fp32 WMMA GEMMs
//   y_n = (a_n + i c_n) y_{n-1} + u_n   (per pole, in V-eigen coordinates)
//   X_right = Y @ V^T                   (WMMA f32 16x16x4)
//   X_left  = X_right - (E@alpha)[:,marks]
// ---------------------------------------------------------------------------

typedef __attribute__((ext_vector_type(2))) float v2f;
typedef __attribute__((ext_vector_type(8))) float v8f;

__device__ __forceinline__ float softplusf(float x) {
  return (x > 20.f) ? x : log1pf(expf(x));
}

// ---------------- V inverse via Gauss-Jordan in LDS (V ~ I, well-conditioned)
__global__ __launch_bounds__(256) void k_inverse(const float* __restrict__ V,
                                                 float* __restrict__ Vinv) {
  __shared__ float M[64][128];
  __shared__ float prow[128];
  __shared__ float fac[64];
  const int tid = threadIdx.x;
  for (int idx = tid; idx < 64 * 128; idx += 256) {
    int r = idx >> 7, c = idx & 127;
    M[r][c] = (c < 64) ? V[r * 64 + c] : ((c - 64 == r) ? 1.f : 0.f);
  }
  __syncthreads();
  for (int k = 0; k < 64; ++k) {
    if (tid < 128) {
      float pinv = 1.f / M[k][k];
      prow[tid] = M[k][tid] * pinv;
    } else if (tid < 192) {
      fac[tid - 128] = M[tid - 128][k];
    }
    __syncthreads();
    for (int idx = tid; idx < 64 * 128; idx += 256) {
      int r = idx >> 7, c = idx & 127;
      M[r][c] = (r == k) ? prow[c] : fmaf(-fac[r], prow[c], M[r][c]);
    }
    __syncthreads();
  }
  for (int idx = tid; idx < 64 * 64; idx += 256) {
    int r = idx >> 6, c = idx & 63;
    Vinv[idx] = M[r][64 + c];
  }
}

// ---------------- G = [gate_w (32x512) ; Vinv@B (64x512)]  (96x512)
__global__ void k_build_G(const float* __restrict__ gate_w,
                          const float* __restrict__ B,
                          const float* __restrict__ Vinv,
                          float* __restrict__ G) {
  int idx = blockIdx.x * blockDim.x + threadIdx.x;
  if (idx >= 96 * 512) return;
  int row = idx >> 9, h = idx & 511;
  if (row < 32) {
    G[idx] = gate_w[row * 512 + h];
  } else {
    int d = row - 32;
    float s = 0.f;
    for (int j = 0; j < 64; ++j) s = fmaf(Vinv[d * 64 + j], B[j * 512 + h], s);
    G[idx] = s;
  }
}

// ---------------- EA = E @ alpha  (64x100)
__global__ void k_EA(const float* __restrict__ E, const float* __restrict__ alpha,
                     float* __restrict__ EA) {
  int idx = blockIdx.x * blockDim.x + threadIdx.x;
  if (idx >= 64 * 100) return;
  int d = idx / 100, k = idx % 100;
  float s = 0.f;
  for (int r = 0; r < 32; ++r) s = fmaf(E[d * 32 + r], alpha[r * 100 + k], s);
  EA[idx] = s;
}

// ---------------- Minv = Vinv @ EA  (64x100)
__global__ void k_Minv(const float* __restrict__ Vinv, const float* __restrict__ EA,
                       float* __restrict__ Minv) {
  int idx = blockIdx.x * blockDim.x + threadIdx.x;
  if (idx >= 64 * 100) return;
  int d = idx / 100, k = idx % 100;
  float s = 0.f;
  for (int j = 0; j < 64; ++j) s = fmaf(Vinv[d * 64 + j], EA[j * 100 + k], s);
  Minv[idx] = s;
}

// ---------------- C96 (N x 96) = U (N x 512) @ G^T  -- WMMA f32 16x16x4
// One wave per 16x16 output tile. A: U rows; B[k][col] = G[col][k].
__global__ __launch_bounds__(32) void k_gemm_uG(const float* __restrict__ U,
                                                const float* __restrict__ G,
                                                float* __restrict__ C96) {
  const int rt = blockIdx.x, ct = blockIdx.y;
  const int l = threadIdx.x;
  const int m = l & 15;            // A row / B col / D col within tile
  const int kh = (l >> 4) * 2;     // K pair selected by lane half
  const float* Arow = U + (size_t)(rt * 16 + m) * 512;
  const float* Brow = G + (size_t)(ct * 16 + m) * 512;
  v8f acc = {};
  for (int kb = 0; kb < 512; kb += 4) {
    v2f a = *(const v2f*)(Arow + kb + kh);
    v2f b = *(const v2f*)(Brow + kb + kh);
    acc = __builtin_amdgcn_wmma_f32_16x16x4_f32(false, a, false, b, (short)0, acc,
                                                false, false);
  }
  const int rbase = rt * 16 + (l >> 4) * 8;
  const int col = ct * 16 + m;
  for (int v = 0; v < 8; ++v) C96[(size_t)(rbase + v) * 96 + col] = acc[v];
}

// ---------------- per-step coefficients: a+ic (decay) and u = q*w + Vinv r
__global__ void k_prep(const float* __restrict__ times, const int* __restrict__ marks,
                       const float* __restrict__ llr, const float* __restrict__ lim,
                       const float* __restrict__ gate_b, const float* __restrict__ C96,
                       const float* __restrict__ Minv, float* __restrict__ Ac,
                       float* __restrict__ Uc, int n) {
  int idx = blockIdx.x * blockDim.x + threadIdx.x;
  if (idx >= n * 32) return;
  int ns = idx >> 5, p = idx & 31;
  float t = times[ns];
  float dt = (ns == 0) ? t : (t - times[ns - 1]);
  float gate = softplusf(C96[(size_t)ns * 96 + p] + gate_b[p]);
  float real = -softplusf(llr[p]) * gate;
  float im = lim[p];
  float e = expf(real * dt);
  float a = e * cosf(im * dt);
  float c = e * sinf(im * dt);
  float denom = real * real + im * im;
  float nre = a - 1.f, nim = c;
  float qr = (nre * real + nim * im) / (denom + 1e-12f);
  float qi = (nim * real - nre * im) / (denom + 1e-12f);
  if (denom < 1e-8f) { qr = dt; qi = 0.f; }
  if (!(dt > 0.f)) { qr = 0.f; qi = 0.f; }  // cont zeroed when dt<=0
  float w0 = C96[(size_t)ns * 96 + 32 + 2 * p];
  float w1 = C96[(size_t)ns * 96 + 32 + 2 * p + 1];
  float ure = qr * w0 - qi * w1;  // Blk(q) acts as complex multiply on pairs
  float uim = qi * w0 + qr * w1;
  int mk = marks[ns];
  ure += Minv[(2 * p) * 100 + mk];
  uim += Minv[(2 * p + 1) * 100 + mk];
  size_t base = (size_t)ns * 64 + 2 * p;
  Ac[base] = a;   Ac[base + 1] = c;
  Uc[base] = ure; Uc[base + 1] = uim;
}

// ---------------- chunked parallel scan (associative: (a,u)∘(A,U)=(aA, aU+u))
__global__ __launch_bounds__(32) void k_scanA(const float* __restrict__ Ac,
                                              const float* __restrict__ Uc,
                                              float* __restrict__ Aagg,
                                              float* __restrict__ Uagg, int n, int L) {
  int ch = blockIdx.x, p = threadIdx.x;
  float Ar = 1.f, Ai = 0.f, Ur = 0.f, Ui = 0.f;
  int i0 = ch * L, i1 = i0 + L; if (i1 > n) i1 = n;
  for (int i = i0; i < i1; ++i) {
    size_t b = (size_t)i * 64 + 2 * p;
    float ar = Ac[b], ai = Ac[b + 1], ur = Uc[b], ui = Uc[b + 1];
    float nAr = ar * Ar - ai * Ai, nAi = ar * Ai + ai * Ar;
    float nUr = ar * Ur - ai * Ui + ur, nUi = ar * Ui + ai * Ur + ui;
    Ar = nAr; Ai = nAi; Ur = nUr; Ui = nUi;
  }
  size_t o = (size_t)ch * 64 + 2 * p;
  Aagg[o] = Ar; Aagg[o + 1] = Ai; Uagg[o] = Ur; Uagg[o + 1] = Ui;
}

__global__ __launch_bounds__(32) void k_scanB(const float* __restrict__ Aagg,
                                              const float* __restrict__ Uagg,
                                              const float* __restrict__ Vinv,
                                              const float* __restrict__ x0,
                                              float* __restrict__ Ystart, int nch) {
  int p = threadIdx.x;
  float yr = 0.f, yi = 0.f;  // y0 = Vinv @ x0
  for (int j = 0; j < 64; ++j) {
    yr = fmaf(Vinv[(2 * p) * 64 + j], x0[j], yr);
    yi = fmaf(Vinv[(2 * p + 1) * 64 + j], x0[j], yi);
  }
  for (int c = 0; c < nch; ++c) {
    size_t o = (size_t)c * 64 + 2 * p;
    Ystart[o] = yr; Ystart[o + 1] = yi;
    float ar = Aagg[o], ai = Aagg[o + 1];
    float nyr = ar * yr - ai * yi + Uagg[o];
    float nyi = ar * yi + ai * yr + Uagg[o + 1];
    yr = nyr; yi = nyi;
  }
}

__global__ __launch_bounds__(32) void k_scanC(const float* __restrict__ Ac,
                                              const float* __restrict__ Uc,
                                              const float* __restrict__ Ystart,
                                              float* __restrict__ Y, int n, int L) {
  int ch = blockIdx.x, p = threadIdx.x;
  size_t so = (size_t)ch * 64 + 2 * p;
  float yr = Ystart[so], yi = Ystart[so + 1];
  int i0 = ch * L, i1 = i0 + L; if (i1 > n) i1 = n;
  for (int i = i0; i < i1; ++i) {
    size_t b = (size_t)i * 64 + 2 * p;
    float ar = Ac[b], ai = Ac[b + 1];
    float nyr = ar * yr - ai * yi + Uc[b];
    float nyi = ar * yi + ai * yr + Uc[b + 1];
    yr = nyr; yi = nyi;
    Y[b] = yr; Y[b + 1] = yi;
  }
}

// ---------------- X_right = Y @ V^T (WMMA), fused X_left = X_right - r epilogue
__global__ __launch_bounds__(32) void k_gemm_out(const float* __restrict__ Y,
                                                 const float* __restrict__ V,
                                                 const float* __restrict__ EA,
                                                 const int* __restrict__ marks,
                                                 float* __restrict__ out, int n) {
  const int rt = blockIdx.x, ct = blockIdx.y, l = threadIdx.x;
  const int m = l & 15;
  const int kh = (l >> 4) * 2;
  const float* Arow = Y + (size_t)(rt * 16 + m) * 64;
  const int dcol = ct * 16 + m;
  const float* Brow = V + (size_t)dcol * 64;  // B[k][d] = V[d][k]
  v8f acc = {};
  for (int kb = 0; kb < 64; kb += 4) {
    v2f a = *(const v2f*)(Arow + kb + kh);
    v2f b = *(const v2f*)(Brow + kb + kh);
    acc = __builtin_amdgcn_wmma_f32_16x16x4_f32(false, a, false, b, (short)0, acc,
                                                false, false);
  }
  const int rbase = rt * 16 + (l >> 4) * 8;
  const size_t xl_off = (size_t)n * 64;
  for (int v = 0; v < 8; ++v) {
    int row = rbase + v;
    float xr = acc[v];
    float r = EA[dcol * 100 + marks[row]];
    out[(size_t)row * 64 + dcol] = xr;            // x_right
    out[xl_off + (size_t)row * 64 + dcol] = xr - r;  // x_left = x_right - r
  }
}

// ---------------------------------------------------------------------------
extern "C" void kernel_launch(void* const* d_in, const int* in_sizes, int n_in,
                              void* d_out, int out_size, void* d_ws, size_t ws_size,
                              hipStream_t stream) {
  const float* times   = (const float*)d_in[0];
  const int*   marks   = (const int*)d_in[1];
  const float* u_in    = (const float*)d_in[2];
  const float* llr     = (const float*)d_in[3];
  const float* lim     = (const float*)d_in[4];
  const float* V       = (const float*)d_in[5];
  const float* B       = (const float*)d_in[6];
  const float* E       = (const float*)d_in[7];
  const float* alpha   = (const float*)d_in[8];
  const float* gate_w  = (const float*)d_in[9];
  const float* gate_b  = (const float*)d_in[10];
  const float* x0      = (const float*)d_in[11];
  float* out = (float*)d_out;
  const int n = in_sizes[0];          // 8192 (multiple of 16)
  const int CHN = 256;                // scan chunks
  const int L = (n + CHN - 1) / CHN;  // steps per chunk (32)

  // workspace carve-up (floats)
  float* ws    = (float*)d_ws;
  float* Vinv  = ws; ws += 64 * 64;
  float* G     = ws; ws += 96 * 512;
  float* EA    = ws; ws += 64 * 100;
  float* Minv  = ws; ws += 64 * 100;
  float* C96   = ws; ws += (size_t)n * 96;
  float* Ac    = ws; ws += (size_t)n * 64;
  float* Uc    = ws; ws += (size_t)n * 64;
  float* Aagg  = ws; ws += (size_t)CHN * 64;
  float* Uagg  = ws; ws += (size_t)CHN * 64;
  float* Ystart= ws; ws += (size_t)CHN * 64;
  float* Y     = ws; ws += (size_t)n * 64;

  k_inverse<<<dim3(1), dim3(256), 0, stream>>>(V, Vinv);
  k_build_G<<<dim3((96 * 512 + 255) / 256), dim3(256), 0, stream>>>(gate_w, B, Vinv, G);
  k_EA<<<dim3((64 * 100 + 255) / 256), dim3(256), 0, stream>>>(E, alpha, EA);
  k_Minv<<<dim3((64 * 100 + 255) / 256), dim3(256), 0, stream>>>(Vinv, EA, Minv);
  k_gemm_uG<<<dim3(n / 16, 6), dim3(32), 0, stream>>>(u_in, G, C96);
  k_prep<<<dim3((n * 32 + 255) / 256), dim3(256), 0, stream>>>(
      times, marks, llr, lim, gate_b, C96, Minv, Ac, Uc, n);
  k_scanA<<<dim3(CHN), dim3(32), 0, stream>>>(Ac, Uc, Aagg, Uagg, n, L);
  k_scanB<<<dim3(1), dim3(32), 0, stream>>>(Aagg, Uagg, Vinv, x0, Ystart, CHN);
  k_scanC<<<dim3(CHN), dim3(32), 0, stream>>>(Ac, Uc, Ystart, Y, n, L);
  k_gemm_out<<<dim3(n / 16, 4), dim3(32), 0, stream>>>(Y, V, EA, marks, out, n);
}